// Augment_60936995995808
// MI455X (gfx1250) — compile-verified
//
#include <hip/hip_runtime.h>
#include <math.h>

// ---------------- problem constants ----------------
#define BATCH   64
#define LEN     131072
#define NFFT    1024
#define HOP     256
#define NT      513            // frames
#define NF      513            // rfft bins
#define BT      (BATCH * NT)   // 32832 rows
#define NCODE   32
#define EPSV    1e-7f
#define LTOT    ((NT - 1) * HOP + NFFT)   // 132096
#define FPAD    544            // NF padded to mult of 32 (K padding for GEMM)
#define NRI     1088           // 2*FPAD : [re | im] packed columns
#define TWO_PI  6.28318530717958647692f

typedef float v2f __attribute__((ext_vector_type(2)));
typedef float v8f __attribute__((ext_vector_type(8)));
typedef unsigned int u32x4 __attribute__((ext_vector_type(4)));
typedef int i32x4 __attribute__((ext_vector_type(4)));
typedef int i32x8 __attribute__((ext_vector_type(8)));

#define TDM_OK __has_builtin(__builtin_amdgcn_tensor_load_to_lds)

__device__ __forceinline__ float hannf(int n) {
    return 0.5f - 0.5f * __cosf(TWO_PI * (float)n / (float)NFFT);
}
// exact twiddle: angle = 2*pi*((a*b) mod NFFT)/NFFT  (int reduction -> full fp32 accuracy)
__device__ __forceinline__ float twc(int a, int b) {
    return cosf(TWO_PI * (float)((a * b) & (NFFT - 1)) / (float)NFFT);
}
__device__ __forceinline__ float tws(int a, int b) {
    return sinf(TWO_PI * (float)((a * b) & (NFFT - 1)) / (float)NFFT);
}

// ---------------- table generation ----------------
// forward DFT matrix [NFFT][NRI]: cols 0..512 = cos, cols 544..1056 = -sin, rest 0
__global__ void gen_dftF(float* __restrict__ m) {
    int i = blockIdx.x * blockDim.x + threadIdx.x;
    if (i >= NFFT * NRI) return;
    int n = i / NRI, c = i % NRI;
    float v = 0.f;
    if (c < NF)                        v =  twc(n, c);
    else if (c >= FPAD && c < FPAD+NF) v = -tws(n, c - FPAD);
    m[i] = v;
}
// autocorr matrix [FPAD][64]: corr[l] = (1/N) * sum_f w_f * P_f * cos(2pi f l / N)
__global__ void gen_acM(float* __restrict__ m) {
    int i = blockIdx.x * blockDim.x + threadIdx.x;
    if (i >= FPAD * 64) return;
    int k = i / 64, l = i % 64;
    float v = 0.f;
    if (k < NF && l <= NCODE) {
        float w = (k == 0 || k == NF - 1) ? 1.f : 2.f;
        v = w * twc(k, l) * (1.f / (float)NFFT);
    }
    m[i] = v;
}
// envelope matrix [64][NRI]: row k = a-coef index; cols: re=cos, im=sin (sign-free magnitude)
__global__ void gen_envM(float* __restrict__ m) {
    int i = blockIdx.x * blockDim.x + threadIdx.x;
    if (i >= 64 * NRI) return;
    int k = i / NRI, c = i % NRI;
    float v = 0.f;
    if (k <= NCODE) {
        if (c < NF)                        v = twc(k, c);
        else if (c >= FPAD && c < FPAD+NF) v = tws(k, c - FPAD);
    }
    m[i] = v;
}
// inverse DFT matrix [NRI][NFFT] with synthesis window folded in
__global__ void gen_idftM(float* __restrict__ m) {
    int i = blockIdx.x * blockDim.x + threadIdx.x;
    if (i >= NRI * NFFT) return;
    int k = i / NFFT, n = i % NFFT;
    float v = 0.f;
    if (k < NF) {
        float w = (k == 0 || k == NF - 1) ? 1.f : 2.f;
        v =  w * twc(k, n) * (1.f / (float)NFFT) * hannf(n);
    } else if (k >= FPAD && k < FPAD + NF) {
        int f = k - FPAD;
        float w = (f == 0 || f == NF - 1) ? 1.f : 2.f;
        v = -w * tws(f, n) * (1.f / (float)NFFT) * hannf(n);
    }
    m[i] = v;
}
// OLA window-square denominator table
__global__ void gen_wsq(float* __restrict__ wsq) {
    int p = blockIdx.x * blockDim.x + threadIdx.x;
    if (p >= LTOT) return;
    float s = 0.f;
    #pragma unroll
    for (int r = 0; r < 4; ++r) {
        int t = p / HOP - r;
        if (t >= 0 && t < NT) { float w = hannf(p - t * HOP); s += w * w; }
    }
    wsq[p] = s;
}

// ---------------- framing (reflect pad + window) ----------------
__global__ void build_frames(const float* __restrict__ wav, float* __restrict__ frames) {
    long long i = (long long)blockIdx.x * blockDim.x + threadIdx.x;
    if (i >= (long long)BT * NFFT) return;
    int n  = (int)(i & (NFFT - 1));
    int bt = (int)(i >> 10);
    int b = bt / NT, t = bt % NT;
    int j = t * HOP + n - NFFT / 2;
    if (j < 0) j = -j;
    if (j >= LEN) j = 2 * LEN - 2 - j;
    frames[i] = wav[(size_t)b * LEN + j] * hannf(n);
}

// ---------------- generic f32 WMMA GEMM: C[M,N] = A[M,K] * B[K,N] ----------------
// REQUIRES: M % 64 == 0, N % 64 == 0, K % 32 == 0 (all call sites satisfy this).
// block tile 64x64, K tile 32; 8 waves; wave = 16 rows x 32 cols (two 16x16 accums).
// A tile staged via the Tensor Data Mover (TDM pads 4 DWORDs per 32 -> LDS stride 36),
// B tile staged transposed so both fragments are contiguous LDS pairs (no repack movs).
__global__ __launch_bounds__(256)
void gemm_wmma_f32(const float* __restrict__ A, const float* __restrict__ Bm,
                   float* __restrict__ C, int M, int N, int K)
{
    __shared__ float As[64][36];   // [m][k], stride 36 (conflict-free: gcd pattern)
    __shared__ float Bs[64][36];   // transposed: [n][k], stride 36
    const int tid  = threadIdx.x;
    const int wid  = tid >> 5;
    const int lane = tid & 31;
    const int bm = blockIdx.x * 64;
    const int bn = blockIdx.y * 64;
    const int wm = (wid & 3) * 16;
    const int wn = (wid >> 2) * 32;
    (void)M;

    v8f acc0 = {}; v8f acc1 = {};

    for (int k0 = 0; k0 < K; k0 += 32) {
#if TDM_OK
        if (wid == 0) {
            // ---- Tensor DMA descriptor (ISA 8.3-8.5), 2D tile: 32 (contig) x 64 rows
            unsigned lds = (unsigned)(unsigned long long)&As[0][0];
            unsigned long long ga = (unsigned long long)(A + (size_t)bm * K + k0);
            u32x4 g0;
            g0[0] = 1u;                                   // count=1, user mode, no gather
            g0[1] = lds;                                  // lds_addr
            g0[2] = (unsigned)ga;                         // global_addr[31:0]
            g0[3] = (unsigned)(ga >> 32) | (2u << 30);    // global_addr[56:32] | type=2
            i32x8 g1;
            g1[0] = (2 << 16)         // data_size = 4B
                  | (1 << 20)         // pad_enable
                  | (4 << 22)         // pad_interval: every 32 DWORDs
                  | (3 << 25);        // pad_amount: 4 DWORDs  -> LDS row stride 36
            g1[1] = (32 & 0xFFFF) << 16;                  // tensor_dim0 = 32 (lo16)
            g1[2] = (64 & 0xFFFF) << 16;                  // dim0 hi | tensor_dim1 = 64 (lo16)
            g1[3] = (32 << 16);                           // dim1 hi | tile_dim0 = 32
            g1[4] = 64;                                   // tile_dim1 = 64, tile_dim2 = 0
            g1[5] = K;                                    // tensor_dim0_stride lo32 (elements)
            g1[6] = 0;                                    // stride0 hi | stride1 lo
            g1[7] = 0;
            i32x4 z4; z4[0] = 0; z4[1] = 0; z4[2] = 0; z4[3] = 0;
            i32x8 z8; z8[0] = 0; z8[1] = 0; z8[2] = 0; z8[3] = 0;
                      z8[4] = 0; z8[5] = 0; z8[6] = 0; z8[7] = 0;
            __builtin_amdgcn_tensor_load_to_lds(g0, g1, z4, z4, z8, 0);
        }
#else
        // manual A staging (vectorized float4)
        #pragma unroll
        for (int q = tid; q < 64 * 8; q += 256) {          // 512 float4
            int r = q >> 3, c4 = (q & 7) << 2;
            const float4 v = *(const float4*)(A + (size_t)(bm + r) * K + k0 + c4);
            As[r][c4 + 0] = v.x; As[r][c4 + 1] = v.y;
            As[r][c4 + 2] = v.z; As[r][c4 + 3] = v.w;
        }
#endif
        // B staging: vector loads of B rows, transposed LDS stores
        #pragma unroll
        for (int q = tid; q < 32 * 16; q += 256) {         // 512 float4
            int r = q >> 4, c4 = (q & 15) << 2;
            const float4 v = *(const float4*)(Bm + (size_t)(k0 + r) * N + bn + c4);
            Bs[c4 + 0][r] = v.x; Bs[c4 + 1][r] = v.y;
            Bs[c4 + 2][r] = v.z; Bs[c4 + 3][r] = v.w;
        }
#if TDM_OK
        if (wid == 0) __builtin_amdgcn_s_wait_tensorcnt(0);
#endif
        __syncthreads();

        const int half = lane >> 4;                        // 0: K pair (0,1); 1: (2,3)
        const int am   = wm + (lane & 15);
        const int bnl  = wn + (lane & 15);
        #pragma unroll
        for (int kk = 0; kk < 32; kk += 4) {
            int koff = kk + half * 2;
            v2f a, b0, b1;
            a.x  = As[am][koff];        a.y  = As[am][koff + 1];
            b0.x = Bs[bnl][koff];       b0.y = Bs[bnl][koff + 1];
            b1.x = Bs[bnl + 16][koff];  b1.y = Bs[bnl + 16][koff + 1];
            acc0 = __builtin_amdgcn_wmma_f32_16x16x4_f32(false, a, false, b0,
                                                         (short)0, acc0, false, false);
            acc1 = __builtin_amdgcn_wmma_f32_16x16x4_f32(false, a, false, b1,
                                                         (short)0, acc1, false, false);
        }
        __syncthreads();
    }
    // C/D layout: VGPR r -> lanes 0-15: (M=r, N=lane); lanes 16-31: (M=r+8, N=lane-16)
    const int mbase = bm + wm + ((lane >> 4) << 3);
    const int gn0   = bn + wn + (lane & 15);
    #pragma unroll
    for (int r = 0; r < 8; ++r) {
        int gm = mbase + r;
        C[(size_t)gm * N + gn0]      = acc0[r];
        C[(size_t)gm * N + gn0 + 16] = acc1[r];
    }
}

// ---------------- norm + power spectrum (one wave per row) ----------------
__global__ __launch_bounds__(256)
void power_kernel(const float* __restrict__ spec, float* __restrict__ power) {
    int wid = threadIdx.x >> 5, lane = threadIdx.x & 31;
    int bt = blockIdx.x * 8 + wid;
    if (bt >= BT) return;
    const float* re = spec + (size_t)bt * NRI;
    const float* im = re + FPAD;
    float mags[17];
    float s = 0.f;
    #pragma unroll
    for (int r = 0; r < 17; ++r) {
        int f = lane + r * 32;
        float m = 0.f;
        if (f < NF) m = sqrtf(re[f] * re[f] + im[f] * im[f]);
        mags[r] = m; s += m;
    }
    #pragma unroll
    for (int off = 16; off > 0; off >>= 1) s += __shfl_xor(s, off, 32);
    float norm = fmaxf(s / (float)NF, EPSV);
    float inv2 = 1.f / (norm * norm);
    #pragma unroll
    for (int r = 0; r < 17; ++r) {
        int f = lane + r * 32;
        if (f < NF) power[(size_t)bt * FPAD + f] = mags[r] * mags[r] * inv2;
    }
    for (int f = NF + lane; f < FPAD; f += 32) power[(size_t)bt * FPAD + f] = 0.f;
}

// ---------------- Levinson-Durbin, order 32 (one row per thread) ----------------
__global__ void levinson_kernel(const float* __restrict__ corr, float* __restrict__ aCoef) {
    int bt = blockIdx.x * blockDim.x + threadIdx.x;
    if (bt >= BT) return;
    float c[NCODE + 1];
    const float* cr = corr + (size_t)bt * 64;
    #pragma unroll
    for (int i = 0; i <= NCODE; ++i) c[i] = cr[i];
    float sol[NCODE + 1], ns[NCODE + 1];
    sol[0] = 1.f;
    sol[1] = -c[1] / fmaxf(c[0], EPSV);
    float extra = c[0] + c[1] * sol[1];
    for (int k = 1; k < NCODE; ++k) {
        float acc = 0.f;
        for (int j = 0; j <= k; ++j) acc += sol[j] * c[k + 1 - j];
        float lam = -acc / fmaxf(extra, EPSV);
        for (int i = 0; i <= k + 1; ++i) {
            float a0 = (i <= k) ? sol[i] : 0.f;
            int ri = k + 1 - i;
            float a1 = (ri <= k) ? sol[ri] : 0.f;
            ns[i] = a0 + lam * a1;
        }
        for (int i = 0; i <= k + 1; ++i) sol[i] = ns[i];
        extra = (1.f - lam * lam) * extra;
    }
    float* a = aCoef + (size_t)bt * 64;
    a[0] = -1.f;                                   // a = [-1, -lpc...]
    #pragma unroll
    for (int i = 1; i <= NCODE; ++i) a[i] = -sol[i];
    #pragma unroll
    for (int i = NCODE + 1; i < 64; ++i) a[i] = 0.f;
}

// ---------------- envelope finish: filt, |source|, angle(source) ----------------
// envRI is overwritten in place with (mag, ang); filt written to its own buffer
__global__ void envfinish_kernel(const float* __restrict__ spec, float* __restrict__ envRI,
                                 float* __restrict__ filt) {
    long long i = (long long)blockIdx.x * blockDim.x + threadIdx.x;
    if (i >= (long long)BT * FPAD) return;
    int bt = (int)(i / FPAD), f = (int)(i % FPAD);
    size_t rix = (size_t)bt * NRI + f;
    float er = envRI[rix], ei = envRI[rix + FPAD];
    float sr = spec[rix],  si = spec[rix + FPAD];
    float den = sqrtf(er * er + ei * ei);
    if (den < EPSV) den = 1.f;
    float fl = 1.f / den;
    float mag = sqrtf(sr * sr + si * si) / (fl + EPSV);
    float ang = atan2f(si, sr);
    bool ok = (f < NF);
    filt[(size_t)bt * FPAD + f] = ok ? fl : 0.f;
    envRI[rix]        = ok ? mag : 0.f;
    envRI[rix + FPAD] = ok ? ang : 0.f;
}

// ---------------- fused freq interpolation + source*filt recombine ----------------
// writes the ISTFT GEMM A-matrix into `dst` (reuses the spec buffer)
__global__ void interp_kernel(const float* __restrict__ filt, const float* __restrict__ envRI,
                              const float* __restrict__ shifts, float* __restrict__ dst) {
    long long i = (long long)blockIdx.x * blockDim.x + threadIdx.x;
    if (i >= (long long)BT * FPAD) return;
    int bt = (int)(i / FPAD), j = (int)(i % FPAD);
    int b = bt / NT;
    float s = shifts[b];
    float re = 0.f, im = 0.f;
    if (j < NF) {
        int lim = (int)floorf((float)NF * s);
        if (j < lim) {
            float pos = ((float)j + 0.5f) / s - 0.5f;
            pos = fminf(fmaxf(pos, 0.f), (float)(NF - 1));
            int i0 = (int)floorf(pos);
            int i1 = min(i0 + 1, NF - 1);
            float w = pos - (float)i0;
            const float* fr = filt + (size_t)bt * FPAD;
            const float* mg = envRI + (size_t)bt * NRI;
            const float* an = mg + FPAD;
            float fI = fr[i0] * (1.f - w) + fr[i1] * w;
            float mI = mg[i0] * (1.f - w) + mg[i1] * w;
            float aI = an[i0] * (1.f - w) + an[i1] * w;
            float amp = mI * fI;
            re = amp * cosf(aI);
            im = amp * sinf(aI);
        }
    }
    size_t rix = (size_t)bt * NRI + j;
    dst[rix] = re;
    dst[rix + FPAD] = im;
}

// ---------------- overlap-add + wsq divide (4-term gather, no atomics) ----------------
__global__ void ola_kernel(const float* __restrict__ oframes, const float* __restrict__ wsq,
                           float* __restrict__ y) {
    long long i = (long long)blockIdx.x * blockDim.x + threadIdx.x;
    if (i >= (long long)BATCH * LEN) return;
    int b = (int)(i / LEN);
    int p = (int)(i % LEN) + NFFT / 2;
    float s = 0.f;
    #pragma unroll
    for (int r = 0; r < 4; ++r) {
        int t = p / HOP - r;
        if (t >= 0 && t < NT) {
            int n = p - t * HOP;
            s += oframes[((size_t)b * NT + t) * NFFT + n];
        }
    }
    y[i] = s / fmaxf(wsq[p], 1e-11f);
}

// ---------------- per-row max, then normalize ----------------
__global__ __launch_bounds__(256)
void rowmax_kernel(const float* __restrict__ y, float* __restrict__ maxv) {
    __shared__ float red[256];
    int b = blockIdx.x, tid = threadIdx.x;
    float m = -3.4e38f;
    for (int i = tid; i < LEN; i += 256) m = fmaxf(m, y[(size_t)b * LEN + i]);
    red[tid] = m;
    __syncthreads();
    for (int s2 = 128; s2 > 0; s2 >>= 1) {
        if (tid < s2) red[tid] = fmaxf(red[tid], red[tid + s2]);
        __syncthreads();
    }
    if (tid == 0) maxv[b] = fmaxf(red[0], EPSV);
}
__global__ void finalize_kernel(const float* __restrict__ y, const float* __restrict__ maxv,
                                float* __restrict__ out) {
    long long i = (long long)blockIdx.x * blockDim.x + threadIdx.x;
    if (i >= (long long)BATCH * LEN) return;
    out[i] = y[i] / maxv[i / LEN];
}

// ---------------- host-side launcher ----------------
extern "C" void kernel_launch(void* const* d_in, const int* in_sizes, int n_in,
                              void* d_out, int out_size, void* d_ws, size_t ws_size,
                              hipStream_t stream) {
    (void)in_sizes; (void)n_in; (void)out_size; (void)ws_size;
    const float* wavs   = (const float*)d_in[0];
    const float* shifts = (const float*)d_in[1];
    float* out = (float*)d_out;

    float* ws = (float*)d_ws;
    size_t off = 0;
    auto alloc = [&](size_t nel) { float* p = ws + off; off += (nel + 63) & ~(size_t)63; return p; };
    float* wsq    = alloc(LTOT);
    float* dftF   = alloc((size_t)NFFT * NRI);
    float* acM    = alloc((size_t)FPAD * 64);
    float* envM   = alloc((size_t)64 * NRI);
    float* idftM  = alloc((size_t)NRI * NFFT);
    float* frames = alloc((size_t)BT * NFFT);   // reused as oframes
    float* spec   = alloc((size_t)BT * NRI);    // reused as srcfilt (ISTFT A-matrix)
    float* power  = alloc((size_t)BT * FPAD);
    float* filt   = alloc((size_t)BT * FPAD);
    float* envRI  = alloc((size_t)BT * NRI);    // reused in place as (mag, ang)
    float* corr   = alloc((size_t)BT * 64);
    float* aCoef  = alloc((size_t)BT * 64);
    float* ybuf   = alloc((size_t)BATCH * LEN);
    float* maxv   = alloc(BATCH);

    const int TPB = 256;
    auto nb = [&](long long n) { return (unsigned)((n + TPB - 1) / TPB); };

    // tables
    gen_wsq  <<<nb(LTOT), TPB, 0, stream>>>(wsq);
    gen_dftF <<<nb((long long)NFFT * NRI), TPB, 0, stream>>>(dftF);
    gen_acM  <<<nb((long long)FPAD * 64), TPB, 0, stream>>>(acM);
    gen_envM <<<nb((long long)64 * NRI), TPB, 0, stream>>>(envM);
    gen_idftM<<<nb((long long)NRI * NFFT), TPB, 0, stream>>>(idftM);

    // framing
    build_frames<<<nb((long long)BT * NFFT), TPB, 0, stream>>>(wavs, frames);

    // STFT:  spec[BT, NRI] = frames[BT,1024] x dftF[1024, NRI]
    gemm_wmma_f32<<<dim3(BT / 64, NRI / 64), TPB, 0, stream>>>(frames, dftF, spec, BT, NRI, NFFT);

    // norm + power spectrum
    power_kernel<<<(BT + 7) / 8, TPB, 0, stream>>>(spec, power);

    // autocorr: corr[BT,64] = power[BT,FPAD] x acM[FPAD,64]
    gemm_wmma_f32<<<dim3(BT / 64, 1), TPB, 0, stream>>>(power, acM, corr, BT, 64, FPAD);

    // Levinson -> a-coefs
    levinson_kernel<<<nb(BT), TPB, 0, stream>>>(corr, aCoef);

    // envelope: envRI[BT,NRI] = aCoef[BT,64] x envM[64,NRI]
    gemm_wmma_f32<<<dim3(BT / 64, NRI / 64), TPB, 0, stream>>>(aCoef, envM, envRI, BT, NRI, 64);

    // filt, |source|, angle(source)
    envfinish_kernel<<<nb((long long)BT * FPAD), TPB, 0, stream>>>(spec, envRI, filt);

    // interpolation + recombine -> srcfilt (reuses spec buffer)
    interp_kernel<<<nb((long long)BT * FPAD), TPB, 0, stream>>>(filt, envRI, shifts, spec);

    // ISTFT: oframes[BT,1024] = srcfilt[BT,NRI] x idftM[NRI,1024]   (window folded in)
    gemm_wmma_f32<<<dim3(BT / 64, NFFT / 64), TPB, 0, stream>>>(spec, idftM, frames, BT, NFFT, NRI);

    // overlap-add + wsq divide
    ola_kernel<<<nb((long long)BATCH * LEN), TPB, 0, stream>>>(frames, wsq, ybuf);

    // peak normalize
    rowmax_kernel<<<BATCH, TPB, 0, stream>>>(ybuf, maxv);
    finalize_kernel<<<nb((long long)BATCH * LEN), TPB, 0, stream>>>(ybuf, maxv, out);
}